// WorldModel_65103114273421
// MI455X (gfx1250) — compile-verified
//
#include <hip/hip_runtime.h>
#include <math.h>

typedef __attribute__((ext_vector_type(2))) float v2f;
typedef __attribute__((ext_vector_type(8))) float v8f;

#define DEV_INLINE __device__ __forceinline__

DEV_INLINE v8f wmma_f32_16x16x4(v2f a, v2f b, v8f c) {
  // D(16x16,f32) = A(16x4,f32) x B(4x16,f32) + C
  return __builtin_amdgcn_wmma_f32_16x16x4_f32(false, a, false, b, (short)0, c,
                                               false, false);
}

static constexpr long long NB = 8192;
static constexpr long long TT = 128;
static constexpr long long BT = NB * TT;  // 1048576 rows
static constexpr long long OFF_MU   = 0;
static constexpr long long OFF_LV   = BT * 16;                 // 16777216
static constexpr long long OFF_REC  = OFF_LV + BT * 16;        // 33554432
static constexpr long long OFF_FLAT = OFF_REC + BT * 2;        // 35651584
static constexpr long long OFF_ROLL = OFF_FLAT + BT * 2;       // 37748736
static constexpr long long OFF_TGT  = OFF_ROLL + NB * 127 * 2; // 39829504

// ---------------------------------------------------------------------------
// Register-resident decoder weights (B-fragments + per-lane biases).
// dec_w3 (16x2) is zero-padded to N=16; lanes with n>=2 hold zeros.
// ---------------------------------------------------------------------------
struct DecW {
  v2f w1[4], w2[4], w3[4];
  float b1, b2, b3;
};

DEV_INLINE DecW load_dec_weights(int lane,
                                 const float* __restrict__ dw1,
                                 const float* __restrict__ db1,
                                 const float* __restrict__ dw2,
                                 const float* __restrict__ db2,
                                 const float* __restrict__ dw3,
                                 const float* __restrict__ db3) {
  DecW W;
  const int half = lane >> 4;
  const int n = lane & 15;
#pragma unroll
  for (int k0 = 0; k0 < 16; k0 += 4) {
    const int k = k0 + 2 * half;
    W.w1[k0 >> 2].x = dw1[k * 16 + n];
    W.w1[k0 >> 2].y = dw1[(k + 1) * 16 + n];
    W.w2[k0 >> 2].x = dw2[k * 16 + n];
    W.w2[k0 >> 2].y = dw2[(k + 1) * 16 + n];
    W.w3[k0 >> 2].x = (n < 2) ? dw3[k * 2 + n] : 0.0f;
    W.w3[k0 >> 2].y = (n < 2) ? dw3[(k + 1) * 2 + n] : 0.0f;
  }
  W.b1 = db1[n];
  W.b2 = db2[n];
  W.b3 = (n < 2) ? db3[n] : 0.0f;
  return W;
}

// ---------------------------------------------------------------------------
// 3-layer decoder on one 16-row tile resident in LDS (weights in registers).
// zb: 16x16 z-tile (stride zs). tb: 16-wide scratch tile (stride ts).
// Returns D-frag of 16x2 output (valid in lanes with n<2).
// ---------------------------------------------------------------------------
DEV_INLINE v8f decoder16(float* zb, int zs, float* tb, int ts, int lane,
                         const DecW& W) {
  const int half = lane >> 4;
  const int n = lane & 15;
  const int m = lane & 15;
  // ---- d1 = relu(z @ dec_w1 + b1)
  v8f c1 = {};
#pragma unroll
  for (int k0 = 0; k0 < 16; k0 += 4) {
    v2f a = *(const v2f*)&zb[m * zs + k0 + 2 * half];
    c1 = wmma_f32_16x16x4(a, W.w1[k0 >> 2], c1);
  }
#pragma unroll
  for (int v = 0; v < 8; ++v)
    tb[(v + 8 * half) * ts + n] = fmaxf(c1[v] + W.b1, 0.0f);
  __syncthreads();
  // ---- read d1 A-frags fully before overwriting tb
  v2f a2[4];
#pragma unroll
  for (int k0 = 0; k0 < 16; k0 += 4)
    a2[k0 >> 2] = *(const v2f*)&tb[m * ts + k0 + 2 * half];
  __syncthreads();
  // ---- d2 = relu(d1 @ dec_w2 + b2)
  v8f c2 = {};
#pragma unroll
  for (int k0 = 0; k0 < 16; k0 += 4)
    c2 = wmma_f32_16x16x4(a2[k0 >> 2], W.w2[k0 >> 2], c2);
#pragma unroll
  for (int v = 0; v < 8; ++v)
    tb[(v + 8 * half) * ts + n] = fmaxf(c2[v] + W.b2, 0.0f);
  __syncthreads();
  // ---- d3 = d2 @ dec_w3 + b3  (B zero-padded to N=16)
  v8f c3 = {};
#pragma unroll
  for (int k0 = 0; k0 < 16; k0 += 4) {
    v2f a = *(const v2f*)&tb[m * ts + k0 + 2 * half];
    c3 = wmma_f32_16x16x4(a, W.w3[k0 >> 2], c3);
  }
#pragma unroll
  for (int v = 0; v < 8; ++v) c3[v] += W.b3;
  return c3;
}

// ---------------------------------------------------------------------------
// Kernel 1: encoder + reparameterization + reconstruction decoder.
// One wave = 16-row tiles, 4 tiles per wave (grid-stride), weights hoisted
// into registers before the tile loop. 2048 blocks x 256 thr (8 waves).
// LDS per wave: bufA 16x36 (layer1 out / z tile), bufB 16x68 (layer2 out).
// ---------------------------------------------------------------------------
__global__ __launch_bounds__(256) void enc_vae_kernel(
    const float* __restrict__ obs, const float* __restrict__ eps,
    const float* __restrict__ ew1, const float* __restrict__ eb1,
    const float* __restrict__ ew2, const float* __restrict__ eb2,
    const float* __restrict__ ew3, const float* __restrict__ eb3,
    const float* __restrict__ dw1, const float* __restrict__ db1,
    const float* __restrict__ dw2, const float* __restrict__ db2,
    const float* __restrict__ dw3, const float* __restrict__ db3,
    float* __restrict__ out) {
  __shared__ float lds[8 * (16 * 36 + 16 * 68)];
  const int wave = threadIdx.x >> 5;
  const int lane = threadIdx.x & 31;
  const int half = lane >> 4;
  const int n = lane & 15;
  const int m = lane & 15;
  float* bufA = lds + wave * (16 * 36 + 16 * 68);
  float* bufB = bufA + 16 * 36;

  // ======== loop-invariant weight fragments (registers) ========
  // layer1: w1 is (2,32); K padded 2->4, lanes of half==1 hold zeros.
  v2f W1[2];
#pragma unroll
  for (int nt = 0; nt < 2; ++nt) {
    const int j = nt * 16 + n;
    W1[nt].x = (half == 0) ? ew1[j] : 0.0f;       // K = 0
    W1[nt].y = (half == 0) ? ew1[32 + j] : 0.0f;  // K = 1
  }
  const float B1[2] = {eb1[n], eb1[16 + n]};
  // layer2: (32x64)
  v2f W2[4][8];
#pragma unroll
  for (int nt = 0; nt < 4; ++nt)
#pragma unroll
    for (int k0 = 0; k0 < 32; k0 += 4) {
      const int k = k0 + 2 * half;
      W2[nt][k0 >> 2].x = ew2[k * 64 + nt * 16 + n];
      W2[nt][k0 >> 2].y = ew2[(k + 1) * 64 + nt * 16 + n];
    }
  const float B2[4] = {eb2[n], eb2[16 + n], eb2[32 + n], eb2[48 + n]};
  // layer3: (64x32) -> mu | log_var halves
  v2f W3[2][16];
#pragma unroll
  for (int nt = 0; nt < 2; ++nt)
#pragma unroll
    for (int k0 = 0; k0 < 64; k0 += 4) {
      const int k = k0 + 2 * half;
      W3[nt][k0 >> 2].x = ew3[k * 32 + nt * 16 + n];
      W3[nt][k0 >> 2].y = ew3[(k + 1) * 32 + nt * 16 + n];
    }
  const float bmu = eb3[n];
  const float blv = eb3[16 + n];
  const DecW DW = load_dec_weights(lane, dw1, db1, dw2, db2, dw3, db3);

  // ======== tile loop: 65536 tiles / 16384 waves = 4 tiles per wave ========
  const long long wgid = (long long)blockIdx.x * 8 + wave;
#pragma unroll 1
  for (int it = 0; it < 4; ++it) {
    const long long tile = wgid + (long long)it * 16384;
    const long long row0 = tile * 16;

    // ---- layer1 via WMMA (K zero-padded): relu(obs @ w1 + b1) -> bufA[16][32]
    v2f aobs;
    {
      v2f o = *(const v2f*)&obs[(row0 + m) * 2];
      aobs.x = (half == 0) ? o.x : 0.0f;
      aobs.y = (half == 0) ? o.y : 0.0f;
    }
#pragma unroll
    for (int nt = 0; nt < 2; ++nt) {
      v8f c = {};
      c = wmma_f32_16x16x4(aobs, W1[nt], c);
#pragma unroll
      for (int v = 0; v < 8; ++v)
        bufA[(v + 8 * half) * 36 + nt * 16 + n] = fmaxf(c[v] + B1[nt], 0.0f);
    }
    __syncthreads();

    // ---- layer2: relu(h1 @ w2 + b2) -> bufB[16][64]; A-frags read once
    v2f A1[8];
#pragma unroll
    for (int k0 = 0; k0 < 32; k0 += 4)
      A1[k0 >> 2] = *(const v2f*)&bufA[m * 36 + k0 + 2 * half];
#pragma unroll
    for (int nt = 0; nt < 4; ++nt) {
      v8f c = {};
#pragma unroll
      for (int k0 = 0; k0 < 32; k0 += 4)
        c = wmma_f32_16x16x4(A1[k0 >> 2], W2[nt][k0 >> 2], c);
#pragma unroll
      for (int v = 0; v < 8; ++v)
        bufB[(v + 8 * half) * 68 + nt * 16 + n] = fmaxf(c[v] + B2[nt], 0.0f);
    }
    __syncthreads();

    // ---- layer3: h2 @ w3 + b3 -> mu | log_var; A-frags read once
    v2f A2[16];
#pragma unroll
    for (int k0 = 0; k0 < 64; k0 += 4)
      A2[k0 >> 2] = *(const v2f*)&bufB[m * 68 + k0 + 2 * half];
    v8f cmu = {}, clv = {};
#pragma unroll
    for (int k0 = 0; k0 < 64; k0 += 4) {
      cmu = wmma_f32_16x16x4(A2[k0 >> 2], W3[0][k0 >> 2], cmu);
      clv = wmma_f32_16x16x4(A2[k0 >> 2], W3[1][k0 >> 2], clv);
    }

    // ---- mu/log_var stores, reparameterize, z tile -> bufA[16][0..15]
#pragma unroll
    for (int v = 0; v < 8; ++v) {
      const long long r = row0 + v + 8 * half;
      const float mu = cmu[v] + bmu;
      const float lv = fminf(fmaxf(clv[v] + blv, -10.0f), 10.0f);
      out[OFF_MU + r * 16 + n] = mu;
      out[OFF_LV + r * 16 + n] = lv;
      const float z = fmaf(expf(0.5f * lv), eps[r * 16 + n], mu);
      bufA[(v + 8 * half) * 36 + n] = z;
    }
    __syncthreads();

    // ---- reconstruction decoder -> obs_recon (16x2)
    v8f d3 = decoder16(bufA, 36, bufB, 68, lane, DW);
    if (n < 2) {
#pragma unroll
      for (int v = 0; v < 8; ++v)
        out[OFF_REC + (row0 + v + 8 * half) * 2 + n] = d3[v];
    }
    __syncthreads();  // tiles reuse bufA/bufB next iteration
  }
}

// ---------------------------------------------------------------------------
// Kernel 2: latent dynamics rollout (127 steps) with fused per-step decode.
// One wave = 16 batch rows. 512 tiles -> 64 blocks x 256 threads.
// All weights (dyn + decoder) register-resident across the t-loop.
// LDS per wave: zb 16x20, tb 16x20.
// ---------------------------------------------------------------------------
__global__ __launch_bounds__(256) void dyn_rollout_kernel(
    const float* __restrict__ act,
    const float* __restrict__ dynA, const float* __restrict__ dynBw,
    const float* __restrict__ dynBb,
    const float* __restrict__ lng, const float* __restrict__ lnbeta,
    const float* __restrict__ dynw, const float* __restrict__ dynb,
    const float* __restrict__ dw1, const float* __restrict__ db1,
    const float* __restrict__ dw2, const float* __restrict__ db2,
    const float* __restrict__ dw3, const float* __restrict__ db3,
    float* __restrict__ out) {
  __shared__ float lds[8 * 2 * 16 * 20];
  const int wave = threadIdx.x >> 5;
  const int lane = threadIdx.x & 31;
  const int half = lane >> 4;
  const int n = lane & 15;
  const int m = lane & 15;
  float* zb = lds + wave * (2 * 16 * 20);
  float* tb = zb + 16 * 20;
  const long long tile = (long long)blockIdx.x * 8 + wave;
  const long long b0 = tile * 16;
  const float* muOut = out + OFF_MU;

  // Register-resident B fragments: dyn_A^T, dyn_w, decoder weights.
  v2f BA[4], BW[4];
#pragma unroll
  for (int k0 = 0; k0 < 16; k0 += 4) {
    const int k = k0 + 2 * half;
    BA[k0 >> 2].x = dynA[n * 16 + k];  // (A^T)[k][n] = A[n][k]
    BA[k0 >> 2].y = dynA[n * 16 + k + 1];
    BW[k0 >> 2].x = dynw[k * 16 + n];
    BW[k0 >> 2].y = dynw[(k + 1) * 16 + n];
  }
  const float bwn = dynBw[n], bbn = dynBb[n];
  const float gn = lng[n], betan = lnbeta[n], dbn = dynb[n];
  const DecW DW = load_dec_weights(lane, dw1, db1, dw2, db2, dw3, db3);

  // z0 = mu[:, 0, :]  (mu row index = batch * 128)
  v8f z;
#pragma unroll
  for (int v = 0; v < 8; ++v) {
    const long long r = (b0 + v + 8 * half) * 128;
    z[v] = muOut[r * 16 + n];
    zb[(v + 8 * half) * 20 + n] = z[v];
  }
  __syncthreads();

#pragma unroll 1
  for (int t = 0; t < 127; ++t) {
    // hh = z @ dyn_A^T + a*dyn_Bw + dyn_Bb
    v8f c = {};
#pragma unroll
    for (int k0 = 0; k0 < 16; k0 += 4) {
      v2f a = *(const v2f*)&zb[m * 20 + k0 + 2 * half];
      c = wmma_f32_16x16x4(a, BA[k0 >> 2], c);
    }
    float hh[8];
#pragma unroll
    for (int v = 0; v < 8; ++v) {
      const float av = act[(b0 + v + 8 * half) * 127 + t];
      hh[v] = fmaf(av, bwn, c[v] + bbn);
    }
    // LayerNorm over the 16 features (features live on 16 lanes of a half).
#pragma unroll
    for (int v = 0; v < 8; ++v) {
      float s = hh[v];
      float s2 = hh[v] * hh[v];
#pragma unroll
      for (int mask = 1; mask < 16; mask <<= 1) {
        s += __shfl_xor(s, mask, 32);
        s2 += __shfl_xor(s2, mask, 32);
      }
      const float mean = s * (1.0f / 16.0f);
      const float var = s2 * (1.0f / 16.0f) - mean * mean;
      const float r = fmaf((hh[v] - mean) * rsqrtf(var + 1e-5f), gn, betan);
      tb[(v + 8 * half) * 20 + n] = fmaxf(r, 0.0f);
    }
    __syncthreads();
    // zn = z + relu(hh_norm) @ dyn_w + dyn_b
    v8f c2 = {};
#pragma unroll
    for (int k0 = 0; k0 < 16; k0 += 4) {
      v2f a = *(const v2f*)&tb[m * 20 + k0 + 2 * half];
      c2 = wmma_f32_16x16x4(a, BW[k0 >> 2], c2);
    }
#pragma unroll
    for (int v = 0; v < 8; ++v) z[v] = z[v] + c2[v] + dbn;
    __syncthreads();  // all reads of old z tile complete
#pragma unroll
    for (int v = 0; v < 8; ++v) zb[(v + 8 * half) * 20 + n] = z[v];
    __syncthreads();
    // fused decode of zn -> rollout_preds[:, t, :]
    v8f d3 = decoder16(zb, 20, tb, 20, lane, DW);
    if (n < 2) {
#pragma unroll
      for (int v = 0; v < 8; ++v)
        out[OFF_ROLL + ((b0 + v + 8 * half) * 127 + t) * 2 + n] = d3[v];
    }
    __syncthreads();  // tb consumed before next iteration overwrites it
  }
}

// ---------------------------------------------------------------------------
// Kernel 3: pure data-movement outputs (obs_flat copy + obs_target slice).
// ---------------------------------------------------------------------------
__global__ __launch_bounds__(256) void copy_outputs_kernel(
    const float* __restrict__ obs, float* __restrict__ out) {
  const long long i = (long long)blockIdx.x * blockDim.x + threadIdx.x;
  if (i < BT * 2) out[OFF_FLAT + i] = obs[i];  // obs_flat
  if (i < NB * 127 * 2) {                      // obs_seq[:,1:,:]
    const long long b = i / 254;
    const long long j = i % 254;  // t*2 + c
    out[OFF_TGT + i] = obs[b * 256 + 2 + j];
  }
}

extern "C" void kernel_launch(void* const* d_in, const int* in_sizes, int n_in,
                              void* d_out, int out_size, void* d_ws,
                              size_t ws_size, hipStream_t stream) {
  const float* obs   = (const float*)d_in[0];
  const float* act   = (const float*)d_in[1];
  const float* eps   = (const float*)d_in[2];
  const float* ew1   = (const float*)d_in[3];
  const float* eb1   = (const float*)d_in[4];
  const float* ew2   = (const float*)d_in[5];
  const float* eb2   = (const float*)d_in[6];
  const float* ew3   = (const float*)d_in[7];
  const float* eb3   = (const float*)d_in[8];
  const float* dw1   = (const float*)d_in[9];
  const float* db1   = (const float*)d_in[10];
  const float* dw2   = (const float*)d_in[11];
  const float* db2   = (const float*)d_in[12];
  const float* dw3   = (const float*)d_in[13];
  const float* db3   = (const float*)d_in[14];
  const float* dynA  = (const float*)d_in[15];
  const float* dynBw = (const float*)d_in[16];
  const float* dynBb = (const float*)d_in[17];
  const float* lng   = (const float*)d_in[18];
  const float* lnb   = (const float*)d_in[19];
  const float* dynw  = (const float*)d_in[20];
  const float* dynb  = (const float*)d_in[21];
  float* out = (float*)d_out;

  // 65536 tiles / (2048 blocks * 8 waves) = 4 tiles per wave
  enc_vae_kernel<<<dim3(2048), dim3(256), 0, stream>>>(
      obs, eps, ew1, eb1, ew2, eb2, ew3, eb3,
      dw1, db1, dw2, db2, dw3, db3, out);

  // 8192 batch rows / 16 per wave / 8 waves = 64 blocks (reads mu from out)
  dyn_rollout_kernel<<<dim3(64), dim3(256), 0, stream>>>(
      act, dynA, dynBw, dynBb, lng, lnb, dynw, dynb,
      dw1, db1, dw2, db2, dw3, db3, out);

  copy_outputs_kernel<<<dim3(8192), dim3(256), 0, stream>>>(obs, out);
}